// WhiteActivation_28406913696441
// MI455X (gfx1250) — compile-verified
//
#include <hip/hip_runtime.h>

// ---------------------------------------------------------------------------
// out[b, f] = white_table[f, idx[b, f]]
// Pure gather / streaming kernel. HBM-bound: 256MB idx in + 256MB out, table
// (1MB) stays resident in L2/LDS. Roofline floor ~22us @ 23.3 TB/s.
//
// CDNA5 strategy:
//   * Stage 64-feature table tile (64KB) into LDS with 16 back-to-back
//     global_load_async_to_lds_b128 ops (ASYNCcnt) + one s_wait_asynccnt.
//   * Random per-element gathers served by LDS (ds_load_b32), not VMEM.
//   * idx/out streamed with non-temporal B128 ops so the 512MB stream does
//     not evict the hot table from L2.
// ---------------------------------------------------------------------------

typedef float v4f __attribute__((ext_vector_type(4)));
typedef int   v4i __attribute__((ext_vector_type(4)));

#define BATCH_N   65536
#define FEAT_N    1024
#define TAB_N     256
#define F_TILE    64          // features per block -> 64KB LDS table tile
#define B_TILE    512         // batch rows per block
#define THREADS_N 256         // 8 wave32s
#define STAGE_IT  ((F_TILE * TAB_N) / 4 / THREADS_N)   // = 16, constant

// Exact pointer types the async builtin expects: <4 x i32> in AS1 / AS3.
typedef __attribute__((address_space(1))) v4i* g_v4i_p;
typedef __attribute__((address_space(3))) v4i* l_v4i_p;

__global__ __launch_bounds__(THREADS_N) void
white_lut_gather(const int* __restrict__ idx,
                 const float* __restrict__ table,
                 float* __restrict__ out)
{
    __shared__ float lut[F_TILE * TAB_N];   // 64KB: [f_local][code]

    const int tid = threadIdx.x;
    const int f0  = blockIdx.x * F_TILE;
    const int b0  = blockIdx.y * B_TILE;

    // ---- stage table tile (contiguous 64KB) into LDS ----
    // Constant trip count (16) so this fully unrolls into straight-line
    // async B128 loads: 256 threads x 16 x 16B = 64KB.
    const float* src = table + (size_t)f0 * TAB_N;
#if __has_builtin(__builtin_amdgcn_global_load_async_to_lds_b128)
    #pragma unroll
    for (int k = 0; k < STAGE_IT; ++k) {
        const int i = tid + k * THREADS_N;      // v4 index
        __builtin_amdgcn_global_load_async_to_lds_b128(
            (g_v4i_p)(src + 4 * i),
            (l_v4i_p)(lut + 4 * i),
            /*offset=*/0, /*cpol=*/0);
    }
  #if __has_builtin(__builtin_amdgcn_s_wait_asynccnt)
    __builtin_amdgcn_s_wait_asynccnt(0);
  #else
    asm volatile("s_wait_asynccnt 0" ::: "memory");
  #endif
#else
    #pragma unroll
    for (int k = 0; k < STAGE_IT; ++k) {
        const int i = tid + k * THREADS_N;
        ((v4f*)lut)[i] = ((const v4f*)src)[i];
    }
#endif
    __syncthreads();

    // ---- gather phase ----
    // lanes 0..15 of each wave cover 64 consecutive features (coalesced 256B),
    // tb dimension walks batch rows.
    const int tf = tid & 15;     // feature quad id: features f0 + 4*tf .. +3
    const int tb = tid >> 4;     // batch slot 0..15, stride 16 rows

    const int fb0 = (tf * 4 + 0) * TAB_N;
    const int fb1 = (tf * 4 + 1) * TAB_N;
    const int fb2 = (tf * 4 + 2) * TAB_N;
    const int fb3 = (tf * 4 + 3) * TAB_N;

    size_t elem = (size_t)(b0 + tb) * FEAT_N + (size_t)f0 + (size_t)tf * 4;
    const v4i* ip = (const v4i*)(idx + elem);
    v4f*       op = (v4f*)(out + elem);
    const int step = (16 * FEAT_N) / 4;   // 16 rows ahead, in v4 units

    #pragma unroll 4
    for (int k = 0; k < B_TILE / 16; ++k) {
        v4i v = __builtin_nontemporal_load(ip);
        v4f r;
        r.x = lut[fb0 + v.x];
        r.y = lut[fb1 + v.y];
        r.z = lut[fb2 + v.z];
        r.w = lut[fb3 + v.w];
        __builtin_nontemporal_store(r, op);
        ip += step;
        op += step;
    }
}

extern "C" void kernel_launch(void* const* d_in, const int* in_sizes, int n_in,
                              void* d_out, int out_size, void* d_ws, size_t ws_size,
                              hipStream_t stream) {
    const int*   idx   = (const int*)d_in[0];     // [BATCH, FEAT] int32 codes
    const float* table = (const float*)d_in[1];   // [FEAT, 256] float32
    float*       out   = (float*)d_out;           // [BATCH, FEAT] float32

    dim3 grid(FEAT_N / F_TILE, BATCH_N / B_TILE); // (16, 128) = 2048 blocks
    white_lut_gather<<<grid, THREADS_N, 0, stream>>>(idx, table, out);
}